// LSTMAutoencoder_4166118277235
// MI455X (gfx1250) — compile-verified
//
#include <hip/hip_runtime.h>
#include <math.h>

// ---------------------------------------------------------------------------
// GRU autoencoder for MI455X (gfx1250, wave32).
//  Phase 1: gi = inputs @ enc_Wih^T + enc_bih        -> fp32 WMMA GEMM
//  Phase 2: sequential GRU recurrence (persistent kernel, LDS-cached Whh
//           slices, per-step grid barrier)           -> VALU/LDS bound
//  Phase 3: gh = states @ dec_Whh^T + dec_bhh        -> fp32 WMMA GEMM
//  Phase 4: elementwise decoder gates
//  Phase 5: out = flip(dec_h @ dec_weight + bias)    -> fp32 WMMA GEMM
//  Phase 6: emb = mean(states, axis=0)
// ---------------------------------------------------------------------------

typedef float v2f __attribute__((ext_vector_type(2)));
typedef float v8f __attribute__((ext_vector_type(8)));

#define TT   8192
#define DD   512
#define HH   1024
#define G3H  3072

// ---------------- fp32 WMMA GEMM:  C[M,N] = A[M,K] * B + bias[N] ------------
// TRANSB=1 : B stored [N,K] (computes A * B^T)   (x @ W^T patterns)
// TRANSB=0 : B stored [K,N]                      (dec_h @ dec_weight)
// FLIP=1   : write row (M-1-m) instead of m      (jnp.flip on output)
template <int TRANSB, int FLIP>
__global__ __launch_bounds__(256)
void wmma_gemm_f32(const float* __restrict__ A, const float* __restrict__ B,
                   const float* __restrict__ bias, float* __restrict__ C,
                   int M, int N, int K)
{
    __shared__ float As[64][17];   // 64 rows x 16 k, pad->conflict free
    __shared__ float Bs[16][65];   // 16 k x 64 n, pad->conflict free

    const int tid  = threadIdx.x;
    const int bn0  = blockIdx.x * 64;
    const int bm0  = blockIdx.y * 64;
    const int wid  = tid >> 5;
    const int lane = tid & 31;
    const int wm   = (wid & 3) * 16;   // wave row offset inside tile
    const int wn   = (wid >> 2) * 32;  // wave col offset inside tile
    const int m16  = lane & 15;
    const int kh   = lane >> 4;        // k-half select (ISA A/B frag layout)

    v8f acc0 = {0.f,0.f,0.f,0.f,0.f,0.f,0.f,0.f};
    v8f acc1 = {0.f,0.f,0.f,0.f,0.f,0.f,0.f,0.f};

    for (int k0 = 0; k0 < K; k0 += 16) {
        {   // stage A tile (64x16), float4 per thread
            int m = tid >> 2, k4 = (tid & 3) * 4;
            const float4 v = *(const float4*)&A[(size_t)(bm0 + m) * K + k0 + k4];
            As[m][k4+0]=v.x; As[m][k4+1]=v.y; As[m][k4+2]=v.z; As[m][k4+3]=v.w;
        }
        if (TRANSB) {   // B[N,K]: thread reads 4 consecutive k of one n-row
            int n = tid >> 2, k4 = (tid & 3) * 4;
            const float4 v = *(const float4*)&B[(size_t)(bn0 + n) * K + k0 + k4];
            Bs[k4+0][n]=v.x; Bs[k4+1][n]=v.y; Bs[k4+2][n]=v.z; Bs[k4+3][n]=v.w;
        } else {        // B[K,N]: thread reads 4 consecutive n of one k-row
            int k = tid >> 4, n4 = (tid & 15) * 4;
            const float4 v = *(const float4*)&B[(size_t)(k0 + k) * N + bn0 + n4];
            Bs[k][n4+0]=v.x; Bs[k][n4+1]=v.y; Bs[k][n4+2]=v.z; Bs[k][n4+3]=v.w;
        }
        __syncthreads();

        #pragma unroll
        for (int kk = 0; kk < 16; kk += 4) {
            const int ka = kk + 2 * kh;
            v2f a;  a.x  = As[wm + m16][ka];       a.y  = As[wm + m16][ka + 1];
            v2f b0; b0.x = Bs[ka][wn + m16];       b0.y = Bs[ka + 1][wn + m16];
            v2f b1; b1.x = Bs[ka][wn + 16 + m16];  b1.y = Bs[ka + 1][wn + 16 + m16];
            acc0 = __builtin_amdgcn_wmma_f32_16x16x4_f32(false, a, false, b0,
                                                         (short)0, acc0, false, false);
            acc1 = __builtin_amdgcn_wmma_f32_16x16x4_f32(false, a, false, b1,
                                                         (short)0, acc1, false, false);
        }
        __syncthreads();
    }

    // C/D layout: VGPR i holds M = i + 8*(lane/16), N = lane%16
    #pragma unroll
    for (int i = 0; i < 8; ++i) {
        const int gm  = bm0 + wm + i + 8 * kh;
        const int gn  = bn0 + wn + m16;
        const int row = FLIP ? (M - 1 - gm) : gm;
        C[(size_t)row * N + gn]      = acc0[i] + bias[gn];
        C[(size_t)row * N + gn + 16] = acc1[i] + bias[gn + 16];
    }
}

// ---------------- persistent encoder recurrence -----------------------------
// 64 blocks x 256 threads; block b owns hidden units [16b, 16b+16).
// LDS: 48 Whh rows (r,z,n per unit) = 192KB + full h vector (4KB) <= 320KB WGP LDS.
#define ENC_BLOCKS 64
#define UNITS_PB   16
#define ENC_LDS_FLOATS (48 * 1024 + 1024)

__global__ __launch_bounds__(256)
void enc_gru_persistent(const float* __restrict__ gi,   // [T,3H], bih folded in
                        const float* __restrict__ Whh,  // [3H,H]
                        const float* __restrict__ bhh,  // [3H]
                        float* __restrict__ enc_states, // [T,H]
                        float* h_cur,                   // [H], zeroed pre-launch
                        unsigned int* bar)              // zeroed pre-launch
{
    extern __shared__ float smem[];
    float* wsm = smem;               // [(u*3+g)*1024 + c]
    float* hs  = smem + 48 * 1024;   // full hidden vector

    const int tid = threadIdx.x;
    const int j0  = blockIdx.x * UNITS_PB;

    // cache this block's 48 weight rows in LDS once
    for (int idx = tid; idx < 48 * 1024; idx += 256) {
        const int u = idx / 3072, rem = idx - u * 3072;
        const int g = rem >> 10, c = rem & 1023;
        wsm[idx] = Whh[(size_t)(g * HH + j0 + u) * HH + c];
    }

    const int u   = tid >> 4;        // unit within block
    const int sub = tid & 15;        // 16 lanes cooperate per unit
    const int c0  = sub * 64;
    const int j   = j0 + u;
    const float* wr = &wsm[(u * 3 + 0) * 1024 + c0];
    const float* wz = &wsm[(u * 3 + 1) * 1024 + c0];
    const float* wn = &wsm[(u * 3 + 2) * 1024 + c0];
    float brj = 0.f, bzj = 0.f, bnj = 0.f;
    if (sub == 0) { brj = bhh[j]; bzj = bhh[HH + j]; bnj = bhh[2 * HH + j]; }
    __syncthreads();

    unsigned int goal = 0;
    for (int t = 0; t < TT; ++t) {
        // pull the cluster-wide hidden state into LDS
        for (int c = tid; c < HH; c += 256)
            hs[c] = ((volatile const float*)h_cur)[c];
        __syncthreads();

        float ar = 0.f, az = 0.f, an = 0.f;
        #pragma unroll 4
        for (int c = 0; c < 64; c += 4) {
            const float4 hv  = *(const float4*)&hs[c0 + c];
            const float4 wrv = *(const float4*)&wr[c];
            const float4 wzv = *(const float4*)&wz[c];
            const float4 wnv = *(const float4*)&wn[c];
            ar += hv.x*wrv.x + hv.y*wrv.y + hv.z*wrv.z + hv.w*wrv.w;
            az += hv.x*wzv.x + hv.y*wzv.y + hv.z*wzv.z + hv.w*wzv.w;
            an += hv.x*wnv.x + hv.y*wnv.y + hv.z*wnv.z + hv.w*wnv.w;
        }
        for (int off = 8; off > 0; off >>= 1) {
            ar += __shfl_down(ar, off, 16);
            az += __shfl_down(az, off, 16);
            an += __shfl_down(an, off, 16);
        }
        if (sub == 0) {
            const float* git = gi + (size_t)t * G3H;
            const float r  = 1.f / (1.f + __expf(-(git[j]          + ar + brj)));
            const float z  = 1.f / (1.f + __expf(-(git[HH + j]     + az + bzj)));
            const float n  = tanhf(git[2 * HH + j] + r * (an + bnj));
            const float hp = hs[j];
            const float hnew = (1.f - z) * n + z * hp;
            enc_states[(size_t)t * HH + j] = hnew;
            h_cur[j] = hnew;
        }
        // grid barrier (monotonic counter; reset by memset before launch)
        __syncthreads();
        goal += gridDim.x;
        if (tid == 0) {
            __threadfence();
            __hip_atomic_fetch_add(bar, 1u, __ATOMIC_ACQ_REL, __HIP_MEMORY_SCOPE_AGENT);
            while (__hip_atomic_load(bar, __ATOMIC_ACQUIRE, __HIP_MEMORY_SCOPE_AGENT) < goal)
                __builtin_amdgcn_s_sleep(1);
        }
        __syncthreads();
    }
}

// ---------------- decoder gates (x = 0 path) --------------------------------
__global__ __launch_bounds__(256)
void dec_gates(const float* __restrict__ gh,       // [T,3H], dec_bhh folded in
               const float* __restrict__ dec_bih,  // [3H]
               const float* __restrict__ enc_states,
               float* __restrict__ dec_h)
{
    const size_t idx = (size_t)blockIdx.x * blockDim.x + threadIdx.x; // T*H
    const int t = (int)(idx >> 10), j = (int)(idx & 1023);
    const float* g = gh + (size_t)t * G3H;
    const float r = 1.f / (1.f + __expf(-(dec_bih[j]          + g[j])));
    const float z = 1.f / (1.f + __expf(-(dec_bih[HH + j]     + g[HH + j])));
    const float n = tanhf(dec_bih[2 * HH + j] + r * g[2 * HH + j]);
    const float hp = enc_states[(size_t)t * HH + j];
    dec_h[(size_t)t * HH + j] = (1.f - z) * n + z * hp;
}

// ---------------- embedding mean --------------------------------------------
__global__ __launch_bounds__(256)
void emb_mean(const float* __restrict__ enc_states, float* __restrict__ out)
{
    const int j = blockIdx.x * blockDim.x + threadIdx.x;  // 1024 lanes total
    float s = 0.f;
    for (int t = 0; t < TT; ++t)
        s += enc_states[(size_t)t * HH + j];              // coalesced rows
    out[j] = s * (1.f / (float)TT);
}

// ---------------------------------------------------------------------------
extern "C" void kernel_launch(void* const* d_in, const int* in_sizes, int n_in,
                              void* d_out, int out_size, void* d_ws, size_t ws_size,
                              hipStream_t stream)
{
    const float* inputs = (const float*)d_in[0];
    const float* eWih   = (const float*)d_in[1];
    const float* eWhh   = (const float*)d_in[2];
    const float* ebih   = (const float*)d_in[3];
    const float* ebhh   = (const float*)d_in[4];
    /* d_in[5] = dec_Wih is dead: decoder input is zero */
    const float* dWhh   = (const float*)d_in[6];
    const float* dbih   = (const float*)d_in[7];
    const float* dbhh   = (const float*)d_in[8];
    const float* dW     = (const float*)d_in[9];
    const float* dbias  = (const float*)d_in[10];
    float* out = (float*)d_out;

    // workspace layout (floats): gi/gh [T*3H] | states [T*H] | dec_h [T*H] | h[H] | bar
    float* ws     = (float*)d_ws;
    float* gi     = ws;                                   // reused as decoder gh
    float* states = gi     + (size_t)TT * G3H;
    float* dech   = states + (size_t)TT * HH;
    float* hcur   = dech   + (size_t)TT * HH;
    unsigned int* bar = (unsigned int*)(hcur + HH);

    // zero h0 + barrier counter (valid graph memset node; replay-safe)
    hipMemsetAsync(hcur, 0, HH * sizeof(float) + 64, stream);

    const dim3 blk(256);

    // 1) gi = inputs @ enc_Wih^T + enc_bih            (M=8192,N=3072,K=512)
    wmma_gemm_f32<1, 0><<<dim3(G3H / 64, TT / 64), blk, 0, stream>>>(
        inputs, eWih, ebih, gi, TT, G3H, DD);

    // 2) sequential recurrence (persistent, 192KB+4KB dynamic LDS per block)
    enc_gru_persistent<<<dim3(ENC_BLOCKS), blk,
                         ENC_LDS_FLOATS * sizeof(float), stream>>>(
        gi, eWhh, ebhh, states, hcur, bar);

    // 3) gh = states @ dec_Whh^T + dec_bhh            (M=8192,N=3072,K=1024)
    wmma_gemm_f32<1, 0><<<dim3(G3H / 64, TT / 64), blk, 0, stream>>>(
        states, dWhh, dbhh, gi, TT, G3H, HH);

    // 4) decoder gates -> dec_h
    dec_gates<<<dim3((TT * HH) / 256), blk, 0, stream>>>(gi, dbih, states, dech);

    // 5) out = flip(dec_h @ dec_weight + dec_bias)    (M=8192,N=512,K=1024)
    wmma_gemm_f32<0, 1><<<dim3(DD / 64, TT / 64), blk, 0, stream>>>(
        dech, dW, dbias, out, TT, DD, HH);

    // 6) emb = mean(states, axis=0)
    emb_mean<<<dim3(HH / 256), blk, 0, stream>>>(states, out + (size_t)TT * DD);
}